// Net_49855980372471
// MI455X (gfx1250) — compile-verified
//
#include <hip/hip_runtime.h>
#include <math.h>

typedef __attribute__((ext_vector_type(2))) float v2f;
typedef __attribute__((ext_vector_type(8))) float v8f;

#define KSZ 5
#define KC  125   // KSZ^3

// ---------------- max pooling (segment max) ----------------
__device__ __forceinline__ unsigned enc_f(float f) {
  unsigned u = __float_as_uint(f);
  return (u & 0x80000000u) ? ~u : (u | 0x80000000u);   // order-preserving map
}
__device__ __forceinline__ float dec_f(unsigned u) {
  unsigned v = (u & 0x80000000u) ? (u & 0x7FFFFFFFu) : ~u;
  return __uint_as_float(v);
}

__global__ void pool_init_k(unsigned* enc, int* cnt, int Nout, int C) {
  int i = blockIdx.x * blockDim.x + threadIdx.x;
  if (i < Nout * C) enc[i] = 0x007FFFFFu;              // enc(-inf)
  if (i < Nout)     cnt[i] = 0;
}

__global__ void pool_scatter_k(const float* __restrict__ x, const int* __restrict__ cluster,
                               unsigned* enc, int* cnt, int Nin, int C) {
  int i = blockIdx.x * blockDim.x + threadIdx.x;
  if (i >= Nin * C) return;
  int n = i / C, c = i - n * C;
  int cl = cluster[n];
  atomicMax(&enc[cl * C + c], enc_f(x[i]));
  if (c == 0) atomicAdd(&cnt[cl], 1);
}

__global__ void pool_final_k(const unsigned* __restrict__ enc, const int* __restrict__ cnt,
                             float* out, int Nout, int C) {
  int i = blockIdx.x * blockDim.x + threadIdx.x;
  if (i >= Nout * C) return;
  int n = i / C;
  out[i] = (cnt[n] > 0) ? dec_f(enc[i]) : 0.0f;
}

// ---------------- spline conv: edge scatter directly into WMMA A-fragment layout ----------
// A-fragment float index for (row r, kdim k):
//   ((r>>4)*Kq + (k>>2))*64 + ((k>>1)&1)*32 + (r&15)*2 + (k&1)
__global__ void edge_scatter_k(const float* __restrict__ x, const int* __restrict__ src,
                               const int* __restrict__ dst, const float* __restrict__ pseudo,
                               float* __restrict__ Afrag, float* deg, int E, int Cin, int Kq) {
  int t = blockIdx.x * blockDim.x + threadIdx.x;
  if (t >= E * Cin) return;
  int e = t / Cin, c = t - e * Cin;
  float fr[3]; int fi[3];
#pragma unroll
  for (int d = 0; d < 3; ++d) {
    float pd = pseudo[e * 3 + d] * (float)(KSZ - 1);
    float fl = floorf(pd);
    fi[d] = (int)fl;
    fr[d] = pd - fl;
  }
  int s = src[e], dn = dst[e];
  float xv = x[s * Cin + c];
  if (c == 0) atomicAdd(&deg[dn], 1.0f);
  size_t baseR = (size_t)(dn >> 4) * Kq * 64 + (size_t)(dn & 15) * 2;
#pragma unroll
  for (int bits = 0; bits < 8; ++bits) {
    float w = 1.0f; int idx = 0;
#pragma unroll
    for (int d = 0; d < 3; ++d) {
      int b = (bits >> d) & 1;
      w *= b ? fr[d] : (1.0f - fr[d]);
      int q = fi[d] + b;
      q = q < 0 ? 0 : (q > KSZ - 1 ? KSZ - 1 : q);
      int pw = (d == 0) ? 1 : (d == 1 ? KSZ : KSZ * KSZ);
      idx += q * pw;
    }
    int k = idx * Cin + c;
    size_t off = baseR + (size_t)(k >> 2) * 64 + (size_t)(((k >> 1) & 1) * 32 + (k & 1));
    atomicAdd(&Afrag[off], w * xv);
  }
}

// ---------------- pack B = W [Kreal, N] into per-lane v2f pairs, zero-padded to 4*Kq ----
__global__ void pack_B_k(const float* __restrict__ B, float* __restrict__ Bp,
                         int Kreal, int Kq, int N) {
  int i = blockIdx.x * blockDim.x + threadIdx.x;       // i = (kq*2+h)*N + c
  int total = Kq * 2 * N;
  if (i >= total) return;
  int c  = i % N;
  int kh = i / N;
  int k0 = (kh >> 1) * 4 + (kh & 1) * 2;
  Bp[2 * i]     = (k0     < Kreal) ? B[(size_t)k0 * N + c]       : 0.0f;
  Bp[2 * i + 1] = (k0 + 1 < Kreal) ? B[(size_t)(k0 + 1) * N + c] : 0.0f;
}

// ---------------- WMMA fp32 GEMM: guard-free inner loop ----------------
// C[M,N] = Afrag(M_pad x 4*Kq, fragment layout) x Bp(packed), NT = N/16.
template <int NT>
__global__ void gemm_wmma_k(const v2f* __restrict__ Af, const v2f* __restrict__ Bf,
                            float* __restrict__ C, int M, int Kq, int N) {
  int lane  = threadIdx.x & 31;
  int wave  = threadIdx.x >> 5;
  int mtile = blockIdx.x * (blockDim.x >> 5) + wave;
  if (mtile * 16 >= M) return;                         // uniform per wave
  int col = lane & 15;
  int h   = lane >> 4;

  v8f acc[NT];
#pragma unroll
  for (int i = 0; i < NT; ++i) acc[i] = (v8f){0.f,0.f,0.f,0.f,0.f,0.f,0.f,0.f};

  const v2f* Ap = Af + (size_t)mtile * Kq * 32 + lane;
  const v2f* Bpp = Bf + (size_t)h * N + col;
  for (int kq = 0; kq < Kq; ++kq) {
    v2f a = Ap[(size_t)kq * 32];
#pragma unroll
    for (int nt = 0; nt < NT; ++nt) {
      v2f b = Bpp[(size_t)kq * 2 * N + nt * 16];
      acc[nt] = __builtin_amdgcn_wmma_f32_16x16x4_f32(
          false, a, false, b, (short)0, acc[nt], false, false);
    }
  }

  int rbase = mtile * 16 + (h << 3);                   // D: row = 8*h + v
#pragma unroll
  for (int nt = 0; nt < NT; ++nt) {
#pragma unroll
    for (int v = 0; v < 8; ++v) {
      int r = rbase + v;
      if (r < M) C[(size_t)r * N + nt * 16 + col] = acc[nt][v];
    }
  }
}

// ---------------- epilogue: out = elu(G/deg + x@root + b) ----------------
__global__ void conv_epilogue_k(const float* __restrict__ G, const float* __restrict__ x,
                                const float* __restrict__ root, const float* __restrict__ bias,
                                const float* __restrict__ deg, float* out,
                                int N, int Cin, int Cout) {
  int i = blockIdx.x * blockDim.x + threadIdx.x;
  if (i >= N * Cout) return;
  int n = i / Cout, o = i - n * Cout;
  float dg = deg[n]; dg = dg > 1.0f ? dg : 1.0f;
  float v = G[i] / dg + bias[o];
  for (int c = 0; c < Cin; ++c) v += x[n * Cin + c] * root[c * Cout + o];
  out[i] = v > 0.0f ? v : expm1f(v);
}

// ---------------- FC head ----------------
__global__ void fc1_k(const float* __restrict__ h, const float* __restrict__ w,
                      const float* __restrict__ b, float* out) {
  int j = blockIdx.x * blockDim.x + threadIdx.x;
  if (j >= 512) return;
  float s = b[j];
  for (int i = 0; i < 1024; ++i) s += h[i] * w[i * 512 + j];
  out[j] = s > 0.0f ? s : expm1f(s);
}

__global__ void fc2_lsm_k(const float* __restrict__ h, const float* __restrict__ w,
                          const float* __restrict__ b, float* out) {
  __shared__ float z[10];
  int o = threadIdx.x;
  if (o < 10) {
    float s = b[o];
    for (int i = 0; i < 512; ++i) s += h[i] * w[i * 10 + o];
    z[o] = s;
  }
  __syncthreads();
  if (o < 10) {
    float m = z[0];
    for (int i = 1; i < 10; ++i) m = fmaxf(m, z[i]);
    float se = 0.0f;
    for (int i = 0; i < 10; ++i) se += expf(z[i] - m);
    out[o] = z[o] - m - logf(se);
  }
}

// ---------------- host orchestration ----------------
static inline int cdiv(int a, int b) { return (a + b - 1) / b; }

extern "C" void kernel_launch(void* const* d_in, const int* in_sizes, int n_in,
                              void* d_out, int out_size, void* d_ws, size_t ws_size,
                              hipStream_t stream) {
  (void)in_sizes; (void)n_in; (void)out_size; (void)ws_size;

  static const int NLEV[6] = {20000, 6000, 2000, 700, 256, 8};
  static const int PREV[6] = {80000, 20000, 6000, 2000, 700, 256};
  static const int EE[5]   = {160000, 48000, 16000, 5600, 2048};
  static const int CIN[5]  = {1, 32, 64, 64, 64};
  static const int COUT[5] = {32, 64, 64, 64, 128};

  const float* x0 = (const float*)d_in[0];
  const int* cluster[6];
  for (int i = 0; i < 6; ++i) cluster[i] = (const int*)d_in[1 + i];
  const int *src[5], *dst[5];
  const float *pseudo[5], *W[5], *root[5], *bias[5];
  for (int i = 0; i < 5; ++i) {
    int base = 7 + i * 6;
    src[i]    = (const int*)  d_in[base + 0];
    dst[i]    = (const int*)  d_in[base + 1];
    pseudo[i] = (const float*)d_in[base + 2];
    W[i]      = (const float*)d_in[base + 3];   // [125, Cin, Cout] == row-major [125*Cin, Cout]
    root[i]   = (const float*)d_in[base + 4];
    bias[i]   = (const float*)d_in[base + 5];
  }
  const float* fc1_w = (const float*)d_in[37];
  const float* fc1_b = (const float*)d_in[38];
  const float* fc2_w = (const float*)d_in[39];
  const float* fc2_b = (const float*)d_in[40];
  float* outp = (float*)d_out;

  // workspace layout (bytes)
  char* ws = (char*)d_ws;
  float*    bufA  = (float*)(ws + 0);              //   640000 f
  float*    bufB  = (float*)(ws + 2560000);        //   640000 f
  float*    G     = (float*)(ws + 5120000);        //   640000 f
  float*    Afrag = (float*)(ws + 7680000);        // 24000000 f (96 MB)
  float*    Bp    = (float*)(ws + 103680000);      //  1024000 f (4 MB)
  float*    deg   = (float*)(ws + 107776000);      //    20000 f
  unsigned* enc   = (unsigned*)(ws + 107856000);   //   640000 u32
  int*      cnt   = (int*)(ws + 110416000);        //    20000 i32
  float*    h1    = (float*)(ws + 110496000);      //      512 f

  // pool helper
  auto pool = [&](const float* in, const int* cl, int Nin, int Nout, int C, float* out) {
    pool_init_k<<<cdiv(Nout * C, 256), 256, 0, stream>>>(enc, cnt, Nout, C);
    pool_scatter_k<<<cdiv(Nin * C, 256), 256, 0, stream>>>(in, cl, enc, cnt, Nin, C);
    pool_final_k<<<cdiv(Nout * C, 256), 256, 0, stream>>>(enc, cnt, out, Nout, C);
  };

  // conv helper
  auto conv = [&](int li, const float* xin, float* xout) {
    int N = NLEV[li], E = EE[li], Cin = CIN[li], Cout = COUT[li];
    int Kreal = KC * Cin;
    int Kq = cdiv(Kreal, 4);                 // K-steps; padded K = 4*Kq
    int Mtiles = cdiv(N, 16);
    hipMemsetAsync(Afrag, 0, (size_t)Mtiles * Kq * 64 * sizeof(float), stream);
    hipMemsetAsync(deg, 0, (size_t)N * sizeof(float), stream);
    edge_scatter_k<<<cdiv(E * Cin, 256), 256, 0, stream>>>(xin, src[li], dst[li], pseudo[li],
                                                           Afrag, deg, E, Cin, Kq);
    pack_B_k<<<cdiv(Kq * 2 * Cout, 256), 256, 0, stream>>>(W[li], Bp, Kreal, Kq, Cout);
    int mblocks = cdiv(Mtiles, 4);
    switch (Cout) {
      case 32:  gemm_wmma_k<2><<<mblocks, 128, 0, stream>>>((const v2f*)Afrag, (const v2f*)Bp,
                                                            G, N, Kq, Cout); break;
      case 64:  gemm_wmma_k<4><<<mblocks, 128, 0, stream>>>((const v2f*)Afrag, (const v2f*)Bp,
                                                            G, N, Kq, Cout); break;
      case 128: gemm_wmma_k<8><<<mblocks, 128, 0, stream>>>((const v2f*)Afrag, (const v2f*)Bp,
                                                            G, N, Kq, Cout); break;
    }
    conv_epilogue_k<<<cdiv(N * Cout, 256), 256, 0, stream>>>(G, xin, root[li], bias[li], deg,
                                                             xout, N, Cin, Cout);
  };

  // pipeline
  pool(x0, cluster[0], PREV[0], NLEV[0], 1, bufA);        // [20000,1]
  conv(0, bufA, bufB);                                    // [20000,32]
  pool(bufB, cluster[1], PREV[1], NLEV[1], 32, bufA);     // [6000,32]
  conv(1, bufA, bufB);                                    // [6000,64]
  pool(bufB, cluster[2], PREV[2], NLEV[2], 64, bufA);     // [2000,64]
  conv(2, bufA, bufB);                                    // [2000,64]
  pool(bufB, cluster[3], PREV[3], NLEV[3], 64, bufA);     // [700,64]
  conv(3, bufA, bufB);                                    // [700,64]
  pool(bufB, cluster[4], PREV[4], NLEV[4], 64, bufA);     // [256,64]
  conv(4, bufA, bufB);                                    // [256,128]
  pool(bufB, cluster[5], PREV[5], NLEV[5], 128, bufA);    // [8,128] -> 1024 contiguous

  fc1_k<<<2, 256, 0, stream>>>(bufA, fc1_w, fc1_b, h1);
  fc2_lsm_k<<<1, 32, 0, stream>>>(h1, fc2_w, fc2_b, outp);
}